// MSDeformableAttention_19988777796187
// MI455X (gfx1250) — compile-verified
//
#include <hip/hip_runtime.h>
#include <hip/hip_bf16.h>

// ---------------------------------------------------------------------------
// MS-Deformable-Attention for MI455X (gfx1250, wave32)
//   bs=16, Lq=1000, EMBED=256, HEADS=8, HEAD_DIM=32, LEVELS=4, POINTS=4
//   Lv = 80*80+40*40+20*20+10*10 = 8500
// Phase 1: WMMA f32 GEMMs (offsets 16000x256x256, attn 16000x256x128)
//          fused with sampling-location epilogue and per-head softmax.
// Phase 2: bilinear gather + weighted sum (lane = channel, coalesced).
// ---------------------------------------------------------------------------

typedef __attribute__((ext_vector_type(2))) float v2f;
typedef __attribute__((ext_vector_type(8))) float v8f;

#define BS 16
#define LQ 1000
#define M_TOTAL (BS * LQ)        // 16000
#define EMBED 256
#define HEADS 8
#define HEAD_DIM 32
#define SUM_POINTS 16
#define N_OFF 256                // HEADS*SUM_POINTS*2
#define N_ATT 128                // HEADS*SUM_POINTS
#define LV 8500

// ---------------------------------------------------------------------------
// Kernel 1: WMMA GEMM + epilogue
//   grid.x = 250 (M super-tiles of 64 rows, 4 waves x 16 rows)
//   grid.y = 24  (tiles 0..15 -> offset GEMM N=256; 16..23 -> attn GEMM N=128)
//   block  = 128 threads (4 waves)
// ---------------------------------------------------------------------------
__global__ __launch_bounds__(128)
void msda_gemm_kernel(const float* __restrict__ query,
                      const float* __restrict__ refpts,   // (bs,Lq,1,4)
                      const float* __restrict__ W_off,    // (256,256)
                      const float* __restrict__ b_off,    // (256)
                      const float* __restrict__ W_attn,   // (256,128)
                      const float* __restrict__ b_attn,   // (128)
                      const float* __restrict__ nps,      // (16) = 0.25
                      float* __restrict__ ws_grids,       // (16000,256)
                      float* __restrict__ ws_attn)        // (16000,128)
{
    const int tileN  = blockIdx.y;
    const bool isAtt = (tileN >= 16);
    const int n0     = isAtt ? (tileN - 16) * 16 : tileN * 16;
    const float* W   = isAtt ? W_attn : W_off;
    const int ldw    = isAtt ? N_ATT : N_OFF;

    const int wave = threadIdx.x >> 5;
    const int lane = threadIdx.x & 31;
    const int half = lane >> 4;          // 0: lanes 0-15, 1: lanes 16-31
    const int col  = lane & 15;
    const int m0   = (blockIdx.x * 4 + wave) * 16;

    // Stage the 256x16 weight tile once per block (16 KB), shared by 4 waves.
    __shared__ float Bs[EMBED * 16];
    for (int i = threadIdx.x; i < EMBED * 16; i += 128) {
        const int k = i >> 4;
        const int n = i & 15;
        Bs[i] = W[k * ldw + n0 + n];
    }
    __syncthreads();

    // A fragment: lane (half,col) holds A[m0+col][k0 + 2*half + {0,1}]
    // B fragment: lane (half,col) holds B[k0 + 2*half + {0,1}][n0+col]
    const float* Arow = query + (size_t)(m0 + col) * EMBED;

    v8f acc = {};
    #pragma unroll 8
    for (int k0 = 0; k0 < EMBED; k0 += 4) {
        const int ka = k0 + 2 * half;
        const float2 af = *reinterpret_cast<const float2*>(Arow + ka);
        v2f a; a.x = af.x; a.y = af.y;
        v2f b; b.x = Bs[ka * 16 + col]; b.y = Bs[(ka + 1) * 16 + col];
        acc = __builtin_amdgcn_wmma_f32_16x16x4_f32(
            /*neg_a=*/false, a, /*neg_b=*/false, b,
            /*c_mod=*/(short)0, acc, /*reuse_a=*/false, /*reuse_b=*/false);
    }

    // D layout: VGPR r -> row = r + 8*half, col = lane&15
    if (!isAtt) {
        // Offset epilogue: grid = 2*(ctr + off*nps*wh*0.5) - 1
        #pragma unroll
        for (int r = 0; r < 8; ++r) {
            const int mm = m0 + r + 8 * half;
            const int nn = n0 + col;
            const int b  = mm / LQ;
            const int q  = mm - b * LQ;
            const int xy = nn & 1;
            const int p  = (nn >> 1) & 15;
            const float* rp = refpts + ((size_t)b * LQ + q) * 4;
            const float off = acc[r] + b_off[nn];
            const float loc = rp[xy] + off * nps[p] * rp[2 + xy] * 0.5f;
            ws_grids[(size_t)mm * N_OFF + nn] = 2.0f * loc - 1.0f;
        }
    } else {
        // Attention epilogue: one tile == one head's 16 points for 16 queries.
        __shared__ float Ls[4][16][17];
        #pragma unroll
        for (int r = 0; r < 8; ++r) {
            Ls[wave][r + 8 * half][col] = acc[r] + b_attn[n0 + col];
        }
        __syncthreads();
        if (lane < 16) {
            const int mm = m0 + lane;
            float mx = -1e30f;
            #pragma unroll
            for (int j = 0; j < 16; ++j) mx = fmaxf(mx, Ls[wave][lane][j]);
            float e[16];
            float s = 0.0f;
            #pragma unroll
            for (int j = 0; j < 16; ++j) {
                e[j] = __expf(Ls[wave][lane][j] - mx);
                s += e[j];
            }
            const float inv = 1.0f / s;
            float* dst = ws_attn + (size_t)mm * N_ATT + n0;
            #pragma unroll
            for (int j = 0; j < 16; ++j) dst[j] = e[j] * inv;
        }
    }
}

// ---------------------------------------------------------------------------
// Kernel 2: bilinear sampling + attention-weighted sum
//   one wave per (b*Lq + q, head); lane = channel (32 lanes = HEAD_DIM)
//   grid = 16000 blocks x 256 threads (8 waves/block) -> 128000 waves
// ---------------------------------------------------------------------------
__device__ __forceinline__ float corner_fetch(const float* __restrict__ vb,
                                              int st, int xi, int yi,
                                              int W, int H)
{
    const bool valid = (xi >= 0) & (xi < W) & (yi >= 0) & (yi < H);
    const int xc = xi < 0 ? 0 : (xi > W - 1 ? W - 1 : xi);
    const int yc = yi < 0 ? 0 : (yi > H - 1 ? H - 1 : yi);
    const float v = vb[(size_t)(st + yc * W + xc) * (HEADS * HEAD_DIM)];
    return valid ? v : 0.0f;
}

__global__ __launch_bounds__(256)
void msda_sample_kernel(const float* __restrict__ value,    // (bs,Lv,8,32)
                        const float* __restrict__ ws_grids, // (16000,256)
                        const float* __restrict__ ws_attn,  // (16000,128)
                        float* __restrict__ out)            // (bs,Lq,256)
{
    const int gwave = blockIdx.x * 8 + (threadIdx.x >> 5);
    const int lane  = threadIdx.x & 31;          // channel
    const int h     = gwave & 7;
    const int m     = gwave >> 3;                // b*Lq + q
    if (m >= M_TOTAL) return;
    const int b = m / LQ;

    const float* g  = ws_grids + (size_t)m * N_OFF + h * (SUM_POINTS * 2);
    const float* aw = ws_attn  + (size_t)m * N_ATT + h * SUM_POINTS;
    const float* vb = value + (((size_t)b * LV) * HEADS + h) * HEAD_DIM + lane;

    const int   Hs[4] = {80, 40, 20, 10};
    const int   Wd[4] = {80, 40, 20, 10};
    const int   st[4] = {0, 6400, 8000, 8400};

    float acc = 0.0f;
    #pragma unroll
    for (int l = 0; l < 4; ++l) {
        const int H = Hs[l], W = Wd[l], s0 = st[l];
        #pragma unroll
        for (int pt = 0; pt < 4; ++pt) {
            const int p = l * 4 + pt;
            const float gx = g[p * 2 + 0];
            const float gy = g[p * 2 + 1];
            const float w  = aw[p];
            const float x  = (gx + 1.0f) * (W * 0.5f) - 0.5f;
            const float y  = (gy + 1.0f) * (H * 0.5f) - 0.5f;
            const float x0f = floorf(x);
            const float y0f = floorf(y);
            const int   x0  = (int)x0f;
            const int   y0  = (int)y0f;
            const float tx  = x - x0f;
            const float ty  = y - y0f;

            const float v00 = corner_fetch(vb, s0, x0,     y0,     W, H);
            const float v10 = corner_fetch(vb, s0, x0 + 1, y0,     W, H);
            const float v01 = corner_fetch(vb, s0, x0,     y0 + 1, W, H);
            const float v11 = corner_fetch(vb, s0, x0 + 1, y0 + 1, W, H);

            const float top = v00 * (1.0f - tx) + v10 * tx;
            const float bot = v01 * (1.0f - tx) + v11 * tx;
            acc += w * (top * (1.0f - ty) + bot * ty);
        }
    }
    out[(size_t)m * EMBED + h * HEAD_DIM + lane] = acc;
}

// ---------------------------------------------------------------------------
extern "C" void kernel_launch(void* const* d_in, const int* in_sizes, int n_in,
                              void* d_out, int out_size, void* d_ws, size_t ws_size,
                              hipStream_t stream)
{
    const float* query  = (const float*)d_in[0];
    const float* refpts = (const float*)d_in[1];
    const float* value  = (const float*)d_in[2];
    const float* W_off  = (const float*)d_in[3];
    const float* b_off  = (const float*)d_in[4];
    const float* W_attn = (const float*)d_in[5];
    const float* b_attn = (const float*)d_in[6];
    const float* nps    = (const float*)d_in[7];
    // d_in[8] = value_spatial_shapes (compile-time constant here)

    float* out      = (float*)d_out;
    float* ws_grids = (float*)d_ws;                        // 16000*256 floats
    float* ws_attn  = ws_grids + (size_t)M_TOTAL * N_OFF;  // 16000*128 floats

    dim3 g1(M_TOTAL / 64, 24);   // 250 x 24
    dim3 b1(128);
    msda_gemm_kernel<<<g1, b1, 0, stream>>>(query, refpts, W_off, b_off,
                                            W_attn, b_attn, nps,
                                            ws_grids, ws_attn);

    dim3 g2(M_TOTAL);            // 16000 blocks x 8 waves = 128000 (b,q,h)
    dim3 b2(256);
    msda_sample_kernel<<<g2, b2, 0, stream>>>(value, ws_grids, ws_attn, out);
}